// EnhancedGATModel_5480378270227
// MI455X (gfx1250) — compile-verified
//
#include <hip/hip_runtime.h>
#include <hip/hip_bf16.h>
#include <math.h>

// ---------------------------------------------------------------------------
// EnhancedGAT (GATv2 x3) for MI455X / gfx1250, wave32.
// Dense projections: V_WMMA_F32_16X16X4_F32, 16x64 macro-tile per wave
// (1 A-fragment feeds 4 WMMAs per K-step; pointer-increment addressing).
// Edge softmax: 3-pass atomic scheme, accumulators L2-resident (192MB L2).
// ---------------------------------------------------------------------------

typedef __attribute__((ext_vector_type(2))) float v2f;
typedef __attribute__((ext_vector_type(8))) float v8f;

#define HC 256          // heads*hid
#define NHEADS 4
#define NEG_SLOPE 0.2f
#define BN_EPS 1e-5f

__device__ __forceinline__ float lrelu(float v) {
    return v > 0.f ? v : NEG_SLOPE * v;
}
// order-preserving float -> uint key (for atomicMax over signed floats)
__device__ __forceinline__ unsigned fkey(float f) {
    unsigned b = __float_as_uint(f);
    return (b & 0x80000000u) ? ~b : (b | 0x80000000u);
}
__device__ __forceinline__ float funkey(unsigned k) {
    unsigned b = (k & 0x80000000u) ? (k & 0x7FFFFFFFu) : ~k;
    return __uint_as_float(b);
}

// ---------------------------------------------------------------------------
// Zero fill
// ---------------------------------------------------------------------------
__global__ void k_zero(float* __restrict__ p, long long n) {
    long long i = (long long)blockIdx.x * blockDim.x + threadIdx.x;
    long long stride = (long long)gridDim.x * blockDim.x;
    for (; i < n; i += stride) p[i] = 0.f;
}

// ---------------------------------------------------------------------------
// WMMA fp32 GEMM: C[M,N] = act(A[M,K] @ B[K,N] + bias).
// One wave -> 16(M) x 64(N) macro-tile = 4 accumulators; K stepped by 4 via
// v_wmma_f32_16x16x4_f32. Requires M%16==0, N%64==0, K%4==0.
// A frag (16x4 fp32): lanes 0-15 hold K=k,k+1; lanes 16-31 hold K=k+2,k+3.
// B frag (4x16 fp32): VGPR0 rows K=k / k+2 (half), VGPR1 rows K=k+1 / k+3.
// ---------------------------------------------------------------------------
__global__ void k_gemm_wmma_f32(const float* __restrict__ A,
                                const float* __restrict__ B,
                                const float* __restrict__ bias,
                                float* __restrict__ C,
                                int M, int N, int K, int do_relu) {
    int w      = (blockIdx.x * blockDim.x + threadIdx.x) >> 5;  // global wave id
    int tilesN = N >> 6;                    // 64-wide macro tiles
    int tiles  = (M >> 4) * tilesN;
    if (w >= tiles) return;                 // wave-uniform: EXEC stays all-1
    int tm = (w / tilesN) << 4;
    int tn = (w % tilesN) << 6;

    int lane = threadIdx.x & 31;
    int half = lane >> 4;                   // 0: K offsets 0,1   1: K offsets 2,3
    int l16  = lane & 15;

    v8f c0 = {}, c1 = {}, c2 = {}, c3 = {};
    const float* Ap = A + (size_t)(tm + l16) * K + 2 * half;
    const float* Bp = B + (size_t)(2 * half) * N + tn + l16;

    for (int k = 0; k < K; k += 4) {
        __builtin_prefetch(Ap + 16, 0, 0);          // global_prefetch_b8
        v2f a;
        a.x = Ap[0];
        a.y = Ap[1];
        v2f b0, b1, b2, b3;
        b0.x = Bp[0];  b0.y = Bp[N];
        b1.x = Bp[16]; b1.y = Bp[N + 16];
        b2.x = Bp[32]; b2.y = Bp[N + 32];
        b3.x = Bp[48]; b3.y = Bp[N + 48];
        c0 = __builtin_amdgcn_wmma_f32_16x16x4_f32(false, a, false, b0, (short)0, c0, false, false);
        c1 = __builtin_amdgcn_wmma_f32_16x16x4_f32(false, a, false, b1, (short)0, c1, false, false);
        c2 = __builtin_amdgcn_wmma_f32_16x16x4_f32(false, a, false, b2, (short)0, c2, false, false);
        c3 = __builtin_amdgcn_wmma_f32_16x16x4_f32(false, a, false, b3, (short)0, c3, false, false);
        Ap += 4;
        Bp += (size_t)4 * N;
    }

    // epilogue: C layout per tile: VGPR r -> row tm + 8*half + r, col = l16
    v8f cc[4] = {c0, c1, c2, c3};
#pragma unroll
    for (int t = 0; t < 4; ++t) {
        int   col = tn + 16 * t + l16;
        float bv  = bias ? bias[col] : 0.f;
        float* Cp = C + (size_t)(tm + 8 * half) * N + col;
#pragma unroll
        for (int r = 0; r < 8; ++r) {
            float v = cc[t][r] + bv;
            if (do_relu) v = fmaxf(v, 0.f);
            Cp[(size_t)r * N] = v;
        }
    }
}

// ---------------------------------------------------------------------------
// Edge pass 1: logits + running segment-max.  Warp per edge; lane handles
// channels {lane, lane+32, ..., lane+224} -> fully coalesced 256-f32 rows.
// head(c) = c>>6, so per-lane partials for j=2h,2h+1 belong to head h.
// ---------------------------------------------------------------------------
__global__ void k_edge_logit_max(const float* __restrict__ xl,
                                 const float* __restrict__ xr,
                                 const float* __restrict__ att,   // flat [4*64]
                                 const int* __restrict__ srcA,
                                 const int* __restrict__ dstA,
                                 int E, int Nn,
                                 float* __restrict__ logit,       // [(E+Nn)*4]
                                 unsigned* __restrict__ mkey) {   // [Nn*4]
    int w = (blockIdx.x * blockDim.x + threadIdx.x) >> 5;
    int Etot = E + Nn;
    if (w >= Etot) return;
    int lane = threadIdx.x & 31;
    int s = (w < E) ? srcA[w] : (w - E);
    int d = (w < E) ? dstA[w] : (w - E);

    const float* pl = xl + (size_t)s * HC;
    const float* pr = xr + (size_t)d * HC;
    float hs[NHEADS] = {0.f, 0.f, 0.f, 0.f};
#pragma unroll
    for (int j = 0; j < 8; ++j) {
        int c = lane + 32 * j;
        float v = lrelu(pl[c] + pr[c]);
        hs[j >> 1] += v * att[c];
    }
#pragma unroll
    for (int off = 16; off; off >>= 1) {
#pragma unroll
        for (int h = 0; h < NHEADS; ++h) hs[h] += __shfl_xor(hs[h], off, 32);
    }
    if (lane == 0) {
#pragma unroll
        for (int h = 0; h < NHEADS; ++h) {
            logit[(size_t)w * NHEADS + h] = hs[h];
            atomicMax(&mkey[(size_t)d * NHEADS + h], fkey(hs[h]));
        }
    }
}

// ---------------------------------------------------------------------------
// Edge pass 2: ex = exp(logit - m[dst]); denom[dst] += ex  (thread/(edge,head))
// ---------------------------------------------------------------------------
__global__ void k_edge_exp_sum(float* __restrict__ logit,
                               const unsigned* __restrict__ mkey,
                               const int* __restrict__ dstA,
                               int E, int Nn,
                               float* __restrict__ den) {
    long long t = (long long)blockIdx.x * blockDim.x + threadIdx.x;
    long long total = (long long)(E + Nn) * NHEADS;
    if (t >= total) return;
    int e = (int)(t >> 2), h = (int)(t & 3);
    int d = (e < E) ? dstA[e] : (e - E);
    float m  = funkey(mkey[(size_t)d * NHEADS + h]);
    float ex = expf(logit[t] - m);
    logit[t] = ex;                              // overwrite with numerator
    atomicAdd(&den[(size_t)d * NHEADS + h], ex);
}

// ---------------------------------------------------------------------------
// Edge pass 3: acc[dst] += ex * xl[src].  Warp per edge, coalesced atomics.
// Prefetch the xl row while the (dependent) ex loads are in flight.
// ---------------------------------------------------------------------------
__global__ void k_edge_aggregate(const float* __restrict__ xl,
                                 const float* __restrict__ logit,  // ex now
                                 const int* __restrict__ srcA,
                                 const int* __restrict__ dstA,
                                 int E, int Nn,
                                 float* __restrict__ acc) {
    int w = (blockIdx.x * blockDim.x + threadIdx.x) >> 5;
    int Etot = E + Nn;
    if (w >= Etot) return;
    int lane = threadIdx.x & 31;
    int s = (w < E) ? srcA[w] : (w - E);
    int d = (w < E) ? dstA[w] : (w - E);

    const float* pl = xl + (size_t)s * HC;
    __builtin_prefetch(pl + lane, 0, 0);        // global_prefetch_b8

    float ex[NHEADS];
#pragma unroll
    for (int h = 0; h < NHEADS; ++h) ex[h] = logit[(size_t)w * NHEADS + h];

    float* pa = acc + (size_t)d * HC;
#pragma unroll
    for (int j = 0; j < 8; ++j) {
        int c = lane + 32 * j;
        atomicAdd(&pa[c], ex[j >> 1] * pl[c]);
    }
}

// ---------------------------------------------------------------------------
// Node finalize: out = relu(BN(acc/den + bias)) [+ residual]
// ---------------------------------------------------------------------------
__global__ void k_node_finalize(const float* __restrict__ acc,
                                const float* __restrict__ den,
                                const float* __restrict__ bias,
                                const float* __restrict__ gamma,
                                const float* __restrict__ beta,
                                const float* __restrict__ mean,
                                const float* __restrict__ var,
                                const float* __restrict__ res,   // nullable
                                float* __restrict__ out, int Nn) {
    long long t = (long long)blockIdx.x * blockDim.x + threadIdx.x;
    if (t >= (long long)Nn * HC) return;
    int n = (int)(t >> 8), c = (int)(t & 255), h = c >> 6;
    float v = acc[t] / den[(size_t)n * NHEADS + h] + bias[c];
    v = gamma[c] * (v - mean[c]) * rsqrtf(var[c] + BN_EPS) + beta[c];
    v = fmaxf(v, 0.f);
    if (res) v += res[t];
    out[t] = v;
}

// ---------------------------------------------------------------------------
// Layer-2 projections: warp per node, xl2/xr2 [N,2] = h[N,256] @ W[256,2]
// ---------------------------------------------------------------------------
__global__ void k_proj2(const float* __restrict__ H,
                        const float* __restrict__ Wl,
                        const float* __restrict__ Wr,
                        float* __restrict__ xl2, float* __restrict__ xr2,
                        int Nn) {
    int n = (blockIdx.x * blockDim.x + threadIdx.x) >> 5;
    if (n >= Nn) return;
    int lane = threadIdx.x & 31;
    const float* hr = H + (size_t)n * HC;
    float a0 = 0.f, a1 = 0.f, b0 = 0.f, b1 = 0.f;
#pragma unroll
    for (int j = 0; j < 8; ++j) {
        int k = lane + 32 * j;
        float hv = hr[k];
        a0 += hv * Wl[k * 2];
        a1 += hv * Wl[k * 2 + 1];
        b0 += hv * Wr[k * 2];
        b1 += hv * Wr[k * 2 + 1];
    }
#pragma unroll
    for (int off = 16; off; off >>= 1) {
        a0 += __shfl_xor(a0, off, 32);
        a1 += __shfl_xor(a1, off, 32);
        b0 += __shfl_xor(b0, off, 32);
        b1 += __shfl_xor(b1, off, 32);
    }
    if (lane == 0) {
        xl2[n * 2] = a0; xl2[n * 2 + 1] = a1;
        xr2[n * 2] = b0; xr2[n * 2 + 1] = b1;
    }
}

// Layer-2 edge pass 1 (1 head, 2 channels): thread per edge
__global__ void k_edge2_logit(const float* __restrict__ xl2,
                              const float* __restrict__ xr2,
                              const float* __restrict__ att2,  // [2]
                              const int* __restrict__ srcA,
                              const int* __restrict__ dstA,
                              int E, int Nn,
                              float* __restrict__ logit,
                              unsigned* __restrict__ mkey) {
    int e = blockIdx.x * blockDim.x + threadIdx.x;
    if (e >= E + Nn) return;
    int s = (e < E) ? srcA[e] : (e - E);
    int d = (e < E) ? dstA[e] : (e - E);
    float lg = att2[0] * lrelu(xl2[s * 2]     + xr2[d * 2]) +
               att2[1] * lrelu(xl2[s * 2 + 1] + xr2[d * 2 + 1]);
    logit[e] = lg;
    atomicMax(&mkey[d], fkey(lg));
}

// Layer-2 edge pass 2: exp, denom, aggregate
__global__ void k_edge2_agg(const float* __restrict__ xl2,
                            const float* __restrict__ logit,
                            const unsigned* __restrict__ mkey,
                            const int* __restrict__ srcA,
                            const int* __restrict__ dstA,
                            int E, int Nn,
                            float* __restrict__ den2,
                            float* __restrict__ acc2) {
    int e = blockIdx.x * blockDim.x + threadIdx.x;
    if (e >= E + Nn) return;
    int s = (e < E) ? srcA[e] : (e - E);
    int d = (e < E) ? dstA[e] : (e - E);
    float ex = expf(logit[e] - funkey(mkey[d]));
    atomicAdd(&den2[d], ex);
    atomicAdd(&acc2[d * 2],     ex * xl2[s * 2]);
    atomicAdd(&acc2[d * 2 + 1], ex * xl2[s * 2 + 1]);
}

// Final: bias + log_softmax over 2 classes
__global__ void k_final(const float* __restrict__ acc2,
                        const float* __restrict__ den2,
                        const float* __restrict__ bias2,
                        float* __restrict__ out, int Nn) {
    int n = blockIdx.x * blockDim.x + threadIdx.x;
    if (n >= Nn) return;
    float inv = 1.f / den2[n];
    float v0 = acc2[n * 2]     * inv + bias2[0];
    float v1 = acc2[n * 2 + 1] * inv + bias2[1];
    float mm = fmaxf(v0, v1);
    float lse = mm + logf(expf(v0 - mm) + expf(v1 - mm));
    out[n * 2]     = v0 - lse;
    out[n * 2 + 1] = v1 - lse;
}

// ---------------------------------------------------------------------------
// Host orchestration
// ---------------------------------------------------------------------------
static inline int cdiv(long long a, int b) { return (int)((a + b - 1) / b); }

extern "C" void kernel_launch(void* const* d_in, const int* in_sizes, int n_in,
                              void* d_out, int out_size, void* d_ws, size_t ws_size,
                              hipStream_t stream) {
    const float* x      = (const float*)d_in[0];
    const int*   eidx   = (const int*)d_in[1];
    const float* W_in   = (const float*)d_in[2];
    const float* b_in   = (const float*)d_in[3];
    const float* Wl0    = (const float*)d_in[4];
    const float* Wr0    = (const float*)d_in[5];
    const float* att0   = (const float*)d_in[6];
    const float* bias0  = (const float*)d_in[7];
    const float* Wl1    = (const float*)d_in[8];
    const float* Wr1    = (const float*)d_in[9];
    const float* att1   = (const float*)d_in[10];
    const float* bias1  = (const float*)d_in[11];
    const float* Wl2    = (const float*)d_in[12];
    const float* Wr2    = (const float*)d_in[13];
    const float* att2   = (const float*)d_in[14];
    const float* bias2  = (const float*)d_in[15];
    const float* bng    = (const float*)d_in[16];
    const float* bnb    = (const float*)d_in[17];
    const float* bnm    = (const float*)d_in[18];
    const float* bnv    = (const float*)d_in[19];
    float* out = (float*)d_out;

    const int Nn = in_sizes[0] / 128;     // 50000
    const int E  = in_sizes[1] / 2;       // 800000
    const int Et = E + Nn;                // with self-loops
    const int* srcA = eidx;
    const int* dstA = eidx + E;

    // workspace carve-up
    float* ws = (float*)d_ws;
    size_t off = 0;
    auto alloc = [&](size_t n) { float* p = ws + off; off += (n + 63) & ~(size_t)63; return p; };
    float*    H0   = alloc((size_t)Nn * 64);
    float*    XL   = alloc((size_t)Nn * HC);
    float*    XR   = alloc((size_t)Nn * HC);
    float*    ACC  = alloc((size_t)Nn * HC);
    float*    H1   = alloc((size_t)Nn * HC);
    float*    H2   = alloc((size_t)Nn * HC);
    unsigned* MK   = (unsigned*)alloc((size_t)Nn * NHEADS);
    float*    DEN  = alloc((size_t)Nn * NHEADS);
    float*    LG   = alloc((size_t)Et * NHEADS);
    float*    XL2  = alloc((size_t)Nn * 2);
    float*    XR2  = alloc((size_t)Nn * 2);
    float*    ACC2 = alloc((size_t)Nn * 2);
    unsigned* MK2  = (unsigned*)alloc((size_t)Nn);
    float*    DEN2 = alloc((size_t)Nn);

    const int TB = 256;
    const int warpsPerBlk = TB / 32;

    // ---- input projection: H0 = relu(x @ W_in + b_in), M=Nn,N=64,K=128 ----
    {
        int tiles = (Nn / 16) * (64 / 64);
        k_gemm_wmma_f32<<<cdiv(tiles, warpsPerBlk), TB, 0, stream>>>(
            x, W_in, b_in, H0, Nn, 64, 128, 1);
    }

    // ---- GAT layers 0 and 1 ----
    for (int layer = 0; layer < 2; ++layer) {
        const float* Hin  = layer == 0 ? H0 : H1;
        int K             = layer == 0 ? 64 : HC;
        const float* Wl   = layer == 0 ? Wl0 : Wl1;
        const float* Wr   = layer == 0 ? Wr0 : Wr1;
        const float* att  = layer == 0 ? att0 : att1;
        const float* bias = layer == 0 ? bias0 : bias1;
        const float* res  = layer == 0 ? nullptr : H1;   // residual only layer 1
        float* Hout       = layer == 0 ? H1 : H2;

        int tiles = (Nn / 16) * (HC / 64);
        k_gemm_wmma_f32<<<cdiv(tiles, warpsPerBlk), TB, 0, stream>>>(
            Hin, Wl, nullptr, XL, Nn, HC, K, 0);
        k_gemm_wmma_f32<<<cdiv(tiles, warpsPerBlk), TB, 0, stream>>>(
            Hin, Wr, nullptr, XR, Nn, HC, K, 0);

        k_zero<<<1024, TB, 0, stream>>>((float*)MK, (long long)Nn * NHEADS);
        k_zero<<<1024, TB, 0, stream>>>(DEN, (long long)Nn * NHEADS);
        k_zero<<<1024, TB, 0, stream>>>(ACC, (long long)Nn * HC);

        k_edge_logit_max<<<cdiv((long long)Et, warpsPerBlk), TB, 0, stream>>>(
            XL, XR, att, srcA, dstA, E, Nn, LG, MK);
        k_edge_exp_sum<<<cdiv((long long)Et * NHEADS, TB), TB, 0, stream>>>(
            LG, MK, dstA, E, Nn, DEN);
        k_edge_aggregate<<<cdiv((long long)Et, warpsPerBlk), TB, 0, stream>>>(
            XL, LG, srcA, dstA, E, Nn, ACC);

        k_node_finalize<<<cdiv((long long)Nn * HC, TB), TB, 0, stream>>>(
            ACC, DEN, bias,
            bng + layer * HC, bnb + layer * HC, bnm + layer * HC, bnv + layer * HC,
            res, Hout, Nn);
    }

    // ---- output layer (1 head, 2 channels) ----
    k_proj2<<<cdiv((long long)Nn, warpsPerBlk), TB, 0, stream>>>(
        H2, Wl2, Wr2, XL2, XR2, Nn);

    k_zero<<<256, TB, 0, stream>>>((float*)MK2, Nn);
    k_zero<<<256, TB, 0, stream>>>(DEN2, Nn);
    k_zero<<<256, TB, 0, stream>>>(ACC2, (long long)Nn * 2);

    k_edge2_logit<<<cdiv((long long)Et, TB), TB, 0, stream>>>(
        XL2, XR2, att2, srcA, dstA, E, Nn, LG, MK2);
    k_edge2_agg<<<cdiv((long long)Et, TB), TB, 0, stream>>>(
        XL2, LG, MK2, srcA, dstA, E, Nn, DEN2, ACC2);
    k_final<<<cdiv((long long)Nn, TB), TB, 0, stream>>>(
        ACC2, DEN2, bias2, out, Nn);
}